// DifferentiableAllocator_26268019982507
// MI455X (gfx1250) — compile-verified
//
#include <hip/hip_runtime.h>

// Sinkhorn OT on 8192x8 cost matrix, 200 iterations.
// Single 1024-thread workgroup; K resident in LDS (256KB of the WGP's 320KB).
// Column reductions via chained V_WMMA_F32_16X16X4_F32 (ones-matrix row-sum).

#define LROWS 8192
#define BCOLS 8
#define NITERS 200
#define NTHR 1024
#define RPT 8            // rows per thread
#define INV_EPS 50.0f    // 1/0.02

typedef __attribute__((ext_vector_type(2))) float v2f;
typedef __attribute__((ext_vector_type(8))) float v8f;

// Sum 32 rows (rows [base, base+32)) of an LDS matrix with STRIDE floats/row,
// 8 valid columns, using 8 chained f32 WMMAs with an all-ones A (16x4).
// A is all ones => D = column-sums of B regardless of K-row<->lane mapping.
// No masking of columns 8..15 is needed: column n of D depends only on
// column n of B, and we discard D columns >= 8 downstream. So every lane
// loads rows[... + (lane&7)] unconditionally (duplicate data, always valid).
// On lanes 0..15, d[0] == column-sum for column (lane), valid for lane < 8.
template <int STRIDE>
__device__ __forceinline__ v8f colsum32_wmma(const float* rows, int base, int lane) {
  const v2f aones = {1.0f, 1.0f};
  const int n = lane & 7;
  const int rsel = (lane < 16) ? 0 : 2;
  // Single base pointer; all 16 loads use compile-time immediate offsets.
  const float* p = rows + (base + rsel) * STRIDE + n;
  float b0[8], b1[8];
  #pragma unroll
  for (int c = 0; c < 8; ++c) {
    b0[c] = p[4 * c * STRIDE];            // row base+rsel+4c
    b1[c] = p[4 * c * STRIDE + STRIDE];   // row base+rsel+4c+1
  }
  v8f d = {};
  #pragma unroll
  for (int c = 0; c < 8; ++c) {
    v2f bv = {b0[c], b1[c]};
    d = __builtin_amdgcn_wmma_f32_16x16x4_f32(false, aones, false, bv,
                                              (short)0, d, false, false);
  }
  return d;
}

// One Sinkhorn iteration. SAVE_F=true (last iteration only) materializes the
// row potentials f_i into frow[] -- keeping them dead in the other 199
// iterations frees 8 VGPRs in the hot loop (eliminates spills).
template <bool SAVE_F>
__device__ __forceinline__ void sinkhorn_iteration(
    const float4* __restrict__ K4, float* __restrict__ Wbuf,
    float* __restrict__ W2, float* __restrict__ gcol,
    const float* __restrict__ lb, const float* __restrict__ loga,
    float* __restrict__ frow, int tid, int lane, int wv) {
  float gr[BCOLS];
  #pragma unroll
  for (int j = 0; j < BCOLS; ++j) gr[j] = gcol[j];

  float T[BCOLS];
  #pragma unroll
  for (int j = 0; j < BCOLS; ++j) T[j] = 0.0f;

  #pragma unroll
  for (int k = 0; k < RPT; ++k) {
    const int r  = tid + (k << 10);
    const int sw = (r >> 3) & 1;
    const float4 q0 = K4[(r << 1) + sw];
    const float4 q1 = K4[(r << 1) + (sw ^ 1)];
    const float v0 = q0.x + gr[0], v1 = q0.y + gr[1];
    const float v2 = q0.z + gr[2], v3 = q0.w + gr[3];
    const float v4 = q1.x + gr[4], v5 = q1.y + gr[5];
    const float v6 = q1.z + gr[6], v7 = q1.w + gr[7];
    const float m = fmaxf(fmaxf(fmaxf(v0, v1), fmaxf(v2, v3)),
                          fmaxf(fmaxf(v4, v5), fmaxf(v6, v7)));
    const float e0 = __expf(v0 - m), e1 = __expf(v1 - m);
    const float e2 = __expf(v2 - m), e3 = __expf(v3 - m);
    const float e4 = __expf(v4 - m), e5 = __expf(v5 - m);
    const float e6 = __expf(v6 - m), e7 = __expf(v7 - m);
    const float S = ((e0 + e1) + (e2 + e3)) + ((e4 + e5) + (e6 + e7));
    const float logS = __logf(S);
    if (SAVE_F) frow[k] = loga[k] - m - logS;   // f_i (row potential)
    const float ri = __expf(loga[k] - logS);    // a_i / S_i  (bounded, safe)
    T[0] += e0 * ri; T[1] += e1 * ri; T[2] += e2 * ri; T[3] += e3 * ri;
    T[4] += e4 * ri; T[5] += e5 * ri; T[6] += e6 * ri; T[7] += e7 * ri;
  }

  // ---- column reduction: 1024 partials -> 8 sums via two WMMA stages ----
  #pragma unroll
  for (int j = 0; j < BCOLS; ++j) Wbuf[tid * 9 + j] = T[j];
  __syncthreads();
  {
    v8f d = colsum32_wmma<9>(Wbuf, wv << 5, lane);  // all 32 waves, EXEC all-1
    if (lane < 16) W2[(wv << 4) + lane] = d[0];     // cols 8..15 garbage, unused
  }
  __syncthreads();
  if (tid < 32) {
    v8f d = colsum32_wmma<16>(W2, 0, lane);
    if (lane < BCOLS) {
      // g_j <- g_j + log b_j - log T_j
      gcol[lane] = gcol[lane] + lb[lane] - __logf(d[0]);
    }
  }
  __syncthreads();
}

__global__ __launch_bounds__(NTHR, 1) void sinkhorn_ot_kernel(
    const float* __restrict__ theta, const float* __restrict__ phi,
    const float* __restrict__ sens_raw, const float* __restrict__ n_raw,
    float* __restrict__ out) {
  // K stored as float4 granules, XOR-swizzled on bit3 of the row index so the
  // two b128 loads per row hit the 2-way bank-conflict floor (not 4-way).
  __shared__ float4 K4[LROWS * 2];       // 256 KB
  __shared__ float  Wbuf[NTHR * 9];      // 36 KB per-thread column partials (pad 9)
  __shared__ float  W2[32 * 16];         // per-wave column partials (stride 16)
  __shared__ float  gcol[BCOLS];         // column potentials g
  __shared__ float  lb[BCOLS];           // log b  (log softmax(phi))
  __shared__ float  redA[32];
  __shared__ float  redB[32];
  __shared__ float  sInv;

  const int tid  = threadIdx.x;
  const int lane = tid & 31;
  const int wv   = tid >> 5;

  // ---------------- init: global sums of n and sens_raw ----------------
  float pn = 0.0f, ps = 0.0f;
  #pragma unroll
  for (int k = 0; k < RPT; ++k) {
    const int r = tid + (k << 10);
    pn += n_raw[r] * 1e5f + 1e3f;
    ps += sens_raw[r];
  }
  #pragma unroll
  for (int off = 16; off > 0; off >>= 1) {
    pn += __shfl_xor(pn, off, 32);
    ps += __shfl_xor(ps, off, 32);
  }
  if (lane == 0) { redA[wv] = pn; redB[wv] = ps; }

  // log_b = log softmax(phi), and g0 = 0  (threads 0..7, redundant tiny work)
  if (tid < BCOLS) {
    float p[BCOLS];
    #pragma unroll
    for (int j = 0; j < BCOLS; ++j) p[j] = phi[j];
    float mx = p[0];
    #pragma unroll
    for (int j = 1; j < BCOLS; ++j) mx = fmaxf(mx, p[j]);
    float s = 0.0f;
    #pragma unroll
    for (int j = 0; j < BCOLS; ++j) s += __expf(p[j] - mx);
    lb[tid]   = p[tid] - mx - __logf(s);
    gcol[tid] = 0.0f;
  }
  __syncthreads();
  if (tid < 32) {
    float a = redA[tid], b = redB[tid];
    #pragma unroll
    for (int off = 16; off > 0; off >>= 1) {
      a += __shfl_xor(a, off, 32);
      b += __shfl_xor(b, off, 32);
    }
    if (tid == 0) { redA[0] = a; redB[0] = b; }
  }
  __syncthreads();
  const float logSumN    = __logf(redA[0]);
  const float invSumSens = 1.0f / (redB[0] + 1e-12f);

  // ---------------- build K = (theta - C) / EPS in LDS; per-row log a ----
  float loga[RPT];
  #pragma unroll
  for (int k = 0; k < RPT; ++k) {
    const int r = tid + (k << 10);
    const float nr = n_raw[r] * 1e5f + 1e3f;
    const float wi = nr * sens_raw[r] * invSumSens;   // n_i * sens_i (normalized)
    loga[k] = __logf(nr) - logSumN;                   // log(n_i / sum n)
    const float4 t0 = ((const float4*)theta)[(r << 1) + 0];
    const float4 t1 = ((const float4*)theta)[(r << 1) + 1];
    float4 k0, k1;
    // err_j = 2^(-2*bits_j), bits = {2,3,4,5,6,8,10,12}
    k0.x = (t0.x - wi * 0x1p-4f)  * INV_EPS;
    k0.y = (t0.y - wi * 0x1p-6f)  * INV_EPS;
    k0.z = (t0.z - wi * 0x1p-8f)  * INV_EPS;
    k0.w = (t0.w - wi * 0x1p-10f) * INV_EPS;
    k1.x = (t1.x - wi * 0x1p-12f) * INV_EPS;
    k1.y = (t1.y - wi * 0x1p-16f) * INV_EPS;
    k1.z = (t1.z - wi * 0x1p-20f) * INV_EPS;
    k1.w = (t1.w - wi * 0x1p-24f) * INV_EPS;
    const int sw = (r >> 3) & 1;
    K4[(r << 1) + sw]       = k0;   // logical granule 0 at slot 0^sw
    K4[(r << 1) + (sw ^ 1)] = k1;   // logical granule 1 at slot 1^sw
  }
  __syncthreads();

  // ---------------- 200 Sinkhorn iterations (last one saves f) ----------
  float frow[RPT];
  for (int it = 0; it < NITERS - 1; ++it) {
    sinkhorn_iteration<false>(K4, Wbuf, W2, gcol, lb, loga, frow, tid, lane, wv);
  }
  sinkhorn_iteration<true>(K4, Wbuf, W2, gcol, lb, loga, frow, tid, lane, wv);

  // ---------------- final P = exp(K + f + g) / sum ----------------
  {
    float gr[BCOLS];
    #pragma unroll
    for (int j = 0; j < BCOLS; ++j) gr[j] = gcol[j];

    float T[BCOLS];
    #pragma unroll
    for (int j = 0; j < BCOLS; ++j) T[j] = 0.0f;

    #pragma unroll
    for (int k = 0; k < RPT; ++k) {
      const int r  = tid + (k << 10);
      const int sw = (r >> 3) & 1;
      float4 q0 = K4[(r << 1) + sw];
      float4 q1 = K4[(r << 1) + (sw ^ 1)];
      const float f = frow[k];
      q0.x = __expf(q0.x + f + gr[0]); q0.y = __expf(q0.y + f + gr[1]);
      q0.z = __expf(q0.z + f + gr[2]); q0.w = __expf(q0.w + f + gr[3]);
      q1.x = __expf(q1.x + f + gr[4]); q1.y = __expf(q1.y + f + gr[5]);
      q1.z = __expf(q1.z + f + gr[6]); q1.w = __expf(q1.w + f + gr[7]);
      T[0] += q0.x; T[1] += q0.y; T[2] += q0.z; T[3] += q0.w;
      T[4] += q1.x; T[5] += q1.y; T[6] += q1.z; T[7] += q1.w;
      K4[(r << 1) + sw]       = q0;   // stash unnormalized P in place of K
      K4[(r << 1) + (sw ^ 1)] = q1;
    }

    #pragma unroll
    for (int j = 0; j < BCOLS; ++j) Wbuf[tid * 9 + j] = T[j];
    __syncthreads();
    {
      v8f d = colsum32_wmma<9>(Wbuf, wv << 5, lane);
      if (lane < 16) W2[(wv << 4) + lane] = d[0];
    }
    __syncthreads();
    if (tid < 32) {
      v8f d = colsum32_wmma<16>(W2, 0, lane);
      float t = (lane < BCOLS) ? d[0] : 0.0f;   // mask duplicated cols 8..15
      t += __shfl_xor(t, 4, 32);
      t += __shfl_xor(t, 2, 32);
      t += __shfl_xor(t, 1, 32);
      if (lane == 0) sInv = 1.0f / (t + 1e-40f);
    }
    __syncthreads();

    const float inv = sInv;
    #pragma unroll
    for (int k = 0; k < RPT; ++k) {
      const int r  = tid + (k << 10);
      const int sw = (r >> 3) & 1;
      float4 p0 = K4[(r << 1) + sw];
      float4 p1 = K4[(r << 1) + (sw ^ 1)];
      p0.x *= inv; p0.y *= inv; p0.z *= inv; p0.w *= inv;
      p1.x *= inv; p1.y *= inv; p1.z *= inv; p1.w *= inv;
      ((float4*)out)[(r << 1) + 0] = p0;
      ((float4*)out)[(r << 1) + 1] = p1;
    }
  }
}

extern "C" void kernel_launch(void* const* d_in, const int* in_sizes, int n_in,
                              void* d_out, int out_size, void* d_ws, size_t ws_size,
                              hipStream_t stream) {
  (void)in_sizes; (void)n_in; (void)out_size; (void)d_ws; (void)ws_size;
  const float* theta    = (const float*)d_in[0];
  const float* phi      = (const float*)d_in[1];
  const float* sens_raw = (const float*)d_in[2];
  const float* n_raw    = (const float*)d_in[3];
  float* out = (float*)d_out;
  sinkhorn_ot_kernel<<<1, NTHR, 0, stream>>>(theta, phi, sens_raw, n_raw, out);
}